// CenterLoss_81123342287605
// MI455X (gfx1250) — compile-verified
//
#include <hip/hip_runtime.h>
#include <math.h>

// ---------------------------------------------------------------------------
// CenterLoss on MI455X (gfx1250, wave32).
// loss = (1/B) * sum_i  ||feat[i] - centers[label[i]]||_2 / bincount(label)[label[i]]
// B=16384, D=512, C=10000. Bandwidth-bound (~60MB @ 23.3TB/s ~ 2.7us).
// Per-sample: one wave, coalesced float4 row loads; intra-wave sum-of-squares
// reduction done with V_WMMA_F32_16X16X4_F32 (B-matrix == all ones, so
// D-row-sums + one shfl_xor(16) give the sum of all 32 lane partials,
// independent of exact WMMA lane layout). Deterministic reduction tree.
// ---------------------------------------------------------------------------

typedef __attribute__((ext_vector_type(2))) float v2f;
typedef __attribute__((ext_vector_type(4))) float v4f;
typedef __attribute__((ext_vector_type(8))) float v8f;

#define FEAT_DIM    512
#define NUM_CLASSES 10000
#define WAVES_PER_BLOCK 8
#define BLOCK_THREADS (WAVES_PER_BLOCK * 32)

// ---- kernel 1: zero the count bins (ws is poisoned, must re-zero per call) --
__global__ void cl_zero_counts(unsigned* __restrict__ cnt, int n) {
    int i = blockIdx.x * blockDim.x + threadIdx.x;
    if (i < n) cnt[i] = 0u;
}

// ---- kernel 2: integer bincount of labels (order-independent atomics) ------
__global__ void cl_bincount(const int* __restrict__ label, unsigned* __restrict__ cnt, int b) {
    int i = blockIdx.x * blockDim.x + threadIdx.x;
    if (i < b) atomicAdd(&cnt[label[i]], 1u);
}

// ---- kernel 3: main — one wave per sample ----------------------------------
__global__ __launch_bounds__(BLOCK_THREADS) void cl_main(
    const float* __restrict__ feat,
    const int*   __restrict__ label,
    const float* __restrict__ centers,
    const unsigned* __restrict__ cnt,
    float* __restrict__ partials)
{
    const int lane = threadIdx.x & 31;
    const int wid  = threadIdx.x >> 5;
    const int i    = blockIdx.x * WAVES_PER_BLOCK + wid;   // sample index

    const int lab = label[i];
    const v4f* __restrict__ f = reinterpret_cast<const v4f*>(feat    + (size_t)i   * FEAT_DIM);
    const v4f* __restrict__ c = reinterpret_cast<const v4f*>(centers + (size_t)lab * FEAT_DIM);

    // 512 floats = 128 float4s per row; lane handles lane + 32*j, j=0..3.
    float acc = 0.0f;
#pragma unroll
    for (int j = 0; j < FEAT_DIM / (32 * 4); ++j) {
        v4f fv = __builtin_nontemporal_load(&f[lane + 32 * j]);  // feat: single use, bypass
        v4f cv = c[lane + 32 * j];                               // centers: keep cached (L2 reuse)
        v4f d  = fv - cv;
        acc = fmaf(d.x, d.x, fmaf(d.y, d.y, fmaf(d.z, d.z, fmaf(d.w, d.w, acc))));
    }

    // ---- wave-wide reduction of 32 lane partials via f32 WMMA --------------
    float tot;
#if __has_builtin(__builtin_amdgcn_wmma_f32_16x16x4_f32)
    {
        v2f a; a[0] = acc;  a[1] = 0.0f;   // each lane's partial occupies one A slot
        v2f b; b[0] = 1.0f; b[1] = 1.0f;   // B == all ones -> D[m][n] = sum_k A[m][k]
        v8f c8 = {};
        v8f d = __builtin_amdgcn_wmma_f32_16x16x4_f32(
            /*neg_a=*/false, a, /*neg_b=*/false, b,
            /*c_mod=*/(short)0, c8, /*reuse_a=*/false, /*reuse_b=*/false);
        // Per lane: sum of its 8 D regs = sum over this lane-half's M rows of
        // sum_k A[m][k]; adding the other half (lane^16) = sum of ALL A slots
        // = sum of all 32 lane partials. Layout-robust since B is uniform 1.
        float h = ((d[0] + d[1]) + (d[2] + d[3])) + ((d[4] + d[5]) + (d[6] + d[7]));
        tot = h + __shfl_xor(h, 16, 32);
    }
#else
    tot = acc;
#pragma unroll
    for (int o = 16; o > 0; o >>= 1) tot += __shfl_xor(tot, o, 32);
#endif

    const float dist    = sqrtf(tot);
    const float contrib = dist / (float)cnt[lab];   // count >= 1 always (includes self)

    // ---- deterministic per-block partial (fixed order) ---------------------
    __shared__ float s_wave[WAVES_PER_BLOCK];
    if (lane == 0) s_wave[wid] = contrib;
    __syncthreads();
    if (threadIdx.x == 0) {
        float p = 0.0f;
#pragma unroll
        for (int j = 0; j < WAVES_PER_BLOCK; ++j) p += s_wave[j];
        partials[blockIdx.x] = p;
    }
}

// ---- kernel 4: deterministic final reduction -------------------------------
__global__ __launch_bounds__(256) void cl_finalize(
    const float* __restrict__ partials, int nparts, float* __restrict__ out, float inv_b)
{
    __shared__ float r[256];
    const int tid = threadIdx.x;
    float t = 0.0f;
    for (int j = tid; j < nparts; j += 256) t += partials[j];  // fixed per-thread order
    r[tid] = t;
    __syncthreads();
#pragma unroll
    for (int s = 128; s > 0; s >>= 1) {
        if (tid < s) r[tid] += r[tid + s];
        __syncthreads();
    }
    if (tid == 0) out[0] = r[0] * inv_b;
}

extern "C" void kernel_launch(void* const* d_in, const int* in_sizes, int n_in,
                              void* d_out, int out_size, void* d_ws, size_t ws_size,
                              hipStream_t stream) {
    const float* feat    = (const float*)d_in[0];
    const int*   label   = (const int*)  d_in[1];
    const float* centers = (const float*)d_in[2];
    float*       out     = (float*)d_out;

    const int B = in_sizes[1];                 // 16384
    const int nblocks = B / WAVES_PER_BLOCK;   // 2048

    // workspace layout: [0, 40000) count bins (u32), then per-block partials
    unsigned* cnt      = (unsigned*)d_ws;
    float*    partials = (float*)((char*)d_ws + ((NUM_CLASSES * sizeof(unsigned) + 255) & ~255));

    cl_zero_counts<<<(NUM_CLASSES + 255) / 256, 256, 0, stream>>>(cnt, NUM_CLASSES);
    cl_bincount  <<<(B + 255) / 256, 256, 0, stream>>>(label, cnt, B);
    cl_main      <<<nblocks, BLOCK_THREADS, 0, stream>>>(feat, label, centers, cnt, partials);
    cl_finalize  <<<1, 256, 0, stream>>>(partials, nblocks, out, 1.0f / (float)B);
}